// SAGE_1099511628225
// MI455X (gfx1250) — compile-verified
//
#include <hip/hip_runtime.h>
#include <math.h>

typedef __attribute__((ext_vector_type(2))) float v2f;
typedef __attribute__((ext_vector_type(8))) float v8f;
typedef __attribute__((ext_vector_type(4))) unsigned int u32x4;
typedef __attribute__((ext_vector_type(8))) int i32x8;
typedef __attribute__((ext_vector_type(4))) int i32x4;

#define DFEAT 128
#define GEMM_WAVES 8   // row tiles per block

// ---------------------------------------------------------------- zero fill
__global__ void zero_f32(float* __restrict__ p, long n) {
  long i = (long)blockIdx.x * blockDim.x + threadIdx.x;
  long stride = (long)gridDim.x * blockDim.x;
  for (; i < n; i += stride) p[i] = 0.0f;
}

// ---------------------------------------------------------------- degrees (3 graphs)
__global__ void deg_kernel(const int* __restrict__ d1, const int* __restrict__ d2,
                           const int* __restrict__ d3, float* __restrict__ deg,
                           int E, int N) {
  int t = blockIdx.x * blockDim.x + threadIdx.x;
  if (t >= 3 * E) return;
  int g = t / E;
  int e = t - g * E;
  const int* dp = (g == 0) ? d1 : ((g == 1) ? d2 : d3);
  atomicAdd(&deg[(size_t)g * N + dp[e]], 1.0f);
}

// ------------------------------------------------- layer0: weighted 3-graph scatter
// S[dst] += (attn[g]/max(deg_g[dst],1)) * X[src]   (one wave per edge, 4 floats/lane)
__global__ void scatter_l0(const float* __restrict__ X,
                           const int* __restrict__ s1, const int* __restrict__ d1,
                           const int* __restrict__ s2, const int* __restrict__ d2,
                           const int* __restrict__ s3, const int* __restrict__ d3,
                           const float* __restrict__ deg,
                           const float* __restrict__ attn,
                           float* __restrict__ S, int E, int N) {
  long wid = ((long)blockIdx.x * blockDim.x + threadIdx.x) >> 5;
  int lane = threadIdx.x & 31;
  if (wid >= 3L * E) return;
  int g = (int)(wid / E);
  long e = wid - (long)g * E;
  const int* sp = (g == 0) ? s1 : ((g == 1) ? s2 : s3);
  const int* dp = (g == 0) ? d1 : ((g == 1) ? d2 : d3);
  int src = sp[e], dst = dp[e];
  float c = attn[g] / fmaxf(deg[(size_t)g * N + dst], 1.0f);
  const float4* xs = (const float4*)(X + (size_t)src * DFEAT);
  float4 f = xs[lane];
  float* sd = S + (size_t)dst * DFEAT + lane * 4;
  atomicAdd(sd + 0, c * f.x);
  atomicAdd(sd + 1, c * f.y);
  atomicAdd(sd + 2, c * f.z);
  atomicAdd(sd + 3, c * f.w);
}

// ------------------------------------------------- layers 1/2: mean scatter (graph1)
__global__ void scatter_mean(const float* __restrict__ X, const int* __restrict__ sp,
                             const int* __restrict__ dp, const float* __restrict__ deg1,
                             float* __restrict__ S, int E) {
  long wid = ((long)blockIdx.x * blockDim.x + threadIdx.x) >> 5;
  int lane = threadIdx.x & 31;
  if (wid >= E) return;
  int src = sp[wid], dst = dp[wid];
  float c = 1.0f / fmaxf(deg1[dst], 1.0f);
  const float4* xs = (const float4*)(X + (size_t)src * DFEAT);
  float4 f = xs[lane];
  float* sd = S + (size_t)dst * DFEAT + lane * 4;
  atomicAdd(sd + 0, c * f.x);
  atomicAdd(sd + 1, c * f.y);
  atomicAdd(sd + 2, c * f.z);
  atomicAdd(sd + 3, c * f.w);
}

// ------------------------------------------------- TDM: 2-D tile (16 cols x 128 rows)
// Loads W[k][col0 .. col0+15] for k = 0..127 into LDS as 16-float rows.
// tensor_dim0 = Hout gives hardware OOB zero-fill for the C=40 layer.
__device__ __forceinline__ void tdm_load_wtile(const float* gaddr, unsigned ldsOff,
                                               unsigned tensor_w, unsigned tensor_h,
                                               unsigned stride_w) {
  unsigned long long ga = (unsigned long long)(uintptr_t)gaddr;
  u32x4 g0;
  g0[0] = 1u;                                   // count=1, user descriptor
  g0[1] = ldsOff;                               // lds_addr (bytes)
  g0[2] = (unsigned)ga;                         // global_addr[31:0]
  g0[3] = ((unsigned)(ga >> 32) & 0x01FFFFFFu)  // global_addr[56:32]
          | (2u << 30);                         // type = 2 ("image")
  const unsigned tile0 = 16u, tile1 = tensor_h; // tile_dim2 = 0 -> 2-D
  i32x8 g1;
  g1[0] = (int)(2u << 16);                                      // data_size = 4B
  g1[1] = (int)((tensor_w & 0xFFFFu) << 16);                    // tensor_dim0 lo16
  g1[2] = (int)((tensor_w >> 16) | ((tensor_h & 0xFFFFu) << 16));
  g1[3] = (int)((tensor_h >> 16) | (tile0 << 16));              // tile_dim0
  g1[4] = (int)(tile1 & 0xFFFFu);                               // tile_dim1
  g1[5] = (int)stride_w;                                        // dim0_stride lo32
  g1[6] = 0;                                                    // stride hi + dim1_stride
  g1[7] = 0;
  i32x4 z4 = {0, 0, 0, 0};
  i32x8 z8 = {0, 0, 0, 0, 0, 0, 0, 0};
  // amdgpu-toolchain (clang-23) 6-arg form
  __builtin_amdgcn_tensor_load_to_lds(g0, g1, z4, z4, z8, 0);
}

// ------------------------------------------------- fused dual GEMM + bias + BN + ReLU
// out = epi( alpha * (As @ Ws) + (An @ Wn) + alpha * bias )
// Block = 8 waves x 32 lanes; block covers 128 rows x 16 cols; weight slices in LDS
// via TENSOR_LOAD_TO_LDS; K fixed = 128; exact f32 WMMA.
__global__ void __launch_bounds__(32 * GEMM_WAVES)
sage_gemm(const float* __restrict__ As, const float* __restrict__ An,
          const float* __restrict__ Ws, const float* __restrict__ Wn,
          const float* __restrict__ bias,
          const float* __restrict__ gam, const float* __restrict__ bet,
          const float* __restrict__ rmean, const float* __restrict__ rvar,
          const float* __restrict__ attn,  // null -> alpha = 1
          float* __restrict__ out, int rows, int Hout, int doBnRelu) {
  __shared__ float ldsB[2][DFEAT * 16];   // [0]=Ws slice, [1]=Wn slice (16 KB)

  const int lane = threadIdx.x;           // 0..31
  const int wv   = threadIdx.y;           // 0..7
  const int half = lane >> 4;             // 0 or 1
  const int l15  = lane & 15;
  const int row0 = (blockIdx.x * GEMM_WAVES + wv) * 16;
  const int col0 = blockIdx.y * 16;
  const int n    = col0 + l15;

  // one wave issues both TDM descriptors, waits, then all waves sync
  if (wv == 0) {
    tdm_load_wtile(Ws + col0, (unsigned)(uintptr_t)&ldsB[0][0],
                   (unsigned)Hout, DFEAT, (unsigned)Hout);
    tdm_load_wtile(Wn + col0, (unsigned)(uintptr_t)&ldsB[1][0],
                   (unsigned)Hout, DFEAT, (unsigned)Hout);
    __builtin_amdgcn_s_wait_tensorcnt(0);
  }
  __syncthreads();

  // A fragment row for this lane (clamped; clamped rows are never stored)
  int ar = row0 + l15; if (ar >= rows) ar = rows - 1;
  const float* ars = As + (size_t)ar * DFEAT;
  const float* arn = An + (size_t)ar * DFEAT;
  const float* lws = &ldsB[0][0];
  const float* lwn = &ldsB[1][0];

  v8f accS = {0.f,0.f,0.f,0.f,0.f,0.f,0.f,0.f};
  v8f accN = {0.f,0.f,0.f,0.f,0.f,0.f,0.f,0.f};

  for (int k = 0; k < DFEAT; k += 4) {
    // ISA 16x4 f32 A layout: lanes 0-15 hold K = k,k+1 ; lanes 16-31 hold K = k+2,k+3
    const int kk = k + half * 2;
    v2f a_s = *(const v2f*)(ars + kk);
    v2f a_n = *(const v2f*)(arn + kk);
    const int bi = kk * 16 + l15;           // B fragment: same K distribution
    v2f bs, bn_;
    bs.x  = lws[bi];  bs.y  = lws[bi + 16];
    bn_.x = lwn[bi];  bn_.y = lwn[bi + 16];
    // EXEC is all-ones here (no divergence in the loop)
    accS = __builtin_amdgcn_wmma_f32_16x16x4_f32(false, a_s, false, bs,
                                                 (short)0, accS, false, false);
    accN = __builtin_amdgcn_wmma_f32_16x16x4_f32(false, a_n, false, bn_,
                                                 (short)0, accN, false, false);
  }

  float alpha = attn ? (attn[0] + attn[1] + attn[2]) : 1.0f;
  if (n < Hout) {
    float bb = bias[n] * alpha;
    float scale = 1.0f, shift = 0.0f;
    if (doBnRelu) {
      scale = gam[n] * rsqrtf(rvar[n] + 1e-5f);
      shift = bet[n] - rmean[n] * scale;
    }
#pragma unroll
    for (int j = 0; j < 8; ++j) {
      // C/D layout: VGPR j -> row (row0+j) for lanes 0-15, (row0+j+8) for lanes 16-31
      int r = row0 + j + half * 8;
      if (r < rows) {
        float v = alpha * accS[j] + accN[j] + bb;
        if (doBnRelu) v = fmaxf(v * scale + shift, 0.0f);
        out[(size_t)r * Hout + n] = v;
      }
    }
  }
}

// ------------------------------------------------- in-place log_softmax, C in (32,64]
__global__ void log_softmax_k(float* __restrict__ Y, int rows, int C) {
  long wid = ((long)blockIdx.x * blockDim.x + threadIdx.x) >> 5;
  int lane = threadIdx.x & 31;
  if (wid >= rows) return;
  float* row = Y + (size_t)wid * C;
  float v0 = row[lane];
  bool has2 = (lane + 32) < C;
  float v1 = has2 ? row[lane + 32] : -3.402823e38f;
  float m = fmaxf(v0, v1);
  for (int off = 16; off > 0; off >>= 1) m = fmaxf(m, __shfl_xor(m, off, 32));
  float s = expf(v0 - m) + (has2 ? expf(v1 - m) : 0.0f);
  for (int off = 16; off > 0; off >>= 1) s += __shfl_xor(s, off, 32);
  float ls = m + logf(s);
  row[lane] = v0 - ls;
  if (has2) row[lane + 32] = v1 - ls;
}

// ----------------------------------------------------------------------------------
extern "C" void kernel_launch(void* const* d_in, const int* in_sizes, int n_in,
                              void* d_out, int out_size, void* d_ws, size_t ws_size,
                              hipStream_t stream) {
  const float* features = (const float*)d_in[0];
  const float* attn     = (const float*)d_in[1];
  const int* src1 = (const int*)d_in[2]; const int* dst1 = (const int*)d_in[3];
  const int* src2 = (const int*)d_in[4]; const int* dst2 = (const int*)d_in[5];
  const int* src3 = (const int*)d_in[6]; const int* dst3 = (const int*)d_in[7];
  const float* Ws0 = (const float*)d_in[8];  const float* Wn0 = (const float*)d_in[9];
  const float* b0  = (const float*)d_in[10];
  const float* Ws1 = (const float*)d_in[11]; const float* Wn1 = (const float*)d_in[12];
  const float* b1  = (const float*)d_in[13];
  const float* Ws2 = (const float*)d_in[14]; const float* Wn2 = (const float*)d_in[15];
  const float* b2  = (const float*)d_in[16];
  const float* g0  = (const float*)d_in[17]; const float* be0 = (const float*)d_in[18];
  const float* rm0 = (const float*)d_in[19]; const float* rv0 = (const float*)d_in[20];
  const float* g1  = (const float*)d_in[21]; const float* be1 = (const float*)d_in[22];
  const float* rm1 = (const float*)d_in[23]; const float* rv1 = (const float*)d_in[24];

  const int N = in_sizes[0] / DFEAT;   // 50000
  const int E = in_sizes[2];           // 600000
  const int C = in_sizes[16];          // 40

  // workspace layout (floats): deg[3N] | S[N*128] | X0[N*128] | AGG[N*128]; X1 aliases S
  float* ws  = (float*)d_ws;
  float* deg = ws;
  float* S   = deg + (size_t)3 * N;
  float* X0  = S + (size_t)N * DFEAT;
  float* AGG = X0 + (size_t)N * DFEAT;
  float* X1  = S;                       // S is dead after layer-0 GEMM
  float* OUT = (float*)d_out;           // N x C logits, then log_softmax in place

  const int rowBlocks = (N + 16 * GEMM_WAVES - 1) / (16 * GEMM_WAVES);
  const dim3 gemmBlk(32, GEMM_WAVES);
  const dim3 gemmGridH(rowBlocks, DFEAT / 16);
  const dim3 gemmGridC(rowBlocks, (C + 15) / 16);
  const long wavesL0 = 3L * E, wavesE = E, wavesR = N;

  // ---- layer 0 ----
  zero_f32<<<2048, 256, 0, stream>>>(deg, 3L * N + (long)N * DFEAT);  // deg + S contiguous
  deg_kernel<<<(3 * E + 255) / 256, 256, 0, stream>>>(dst1, dst2, dst3, deg, E, N);
  scatter_l0<<<(int)((wavesL0 * 32 + 255) / 256), 256, 0, stream>>>(
      features, src1, dst1, src2, dst2, src3, dst3, deg, attn, S, E, N);
  sage_gemm<<<gemmGridH, gemmBlk, 0, stream>>>(features, S, Ws0, Wn0, b0,
                                               g0, be0, rm0, rv0, attn, X0, N, DFEAT, 1);

  // ---- layer 1 ----
  zero_f32<<<2048, 256, 0, stream>>>(AGG, (long)N * DFEAT);
  scatter_mean<<<(int)((wavesE * 32 + 255) / 256), 256, 0, stream>>>(X0, src1, dst1, deg, AGG, E);
  sage_gemm<<<gemmGridH, gemmBlk, 0, stream>>>(X0, AGG, Ws1, Wn1, b1,
                                               g1, be1, rm1, rv1, nullptr, X1, N, DFEAT, 1);

  // ---- layer 2 ----
  zero_f32<<<2048, 256, 0, stream>>>(AGG, (long)N * DFEAT);
  scatter_mean<<<(int)((wavesE * 32 + 255) / 256), 256, 0, stream>>>(X1, src1, dst1, deg, AGG, E);
  sage_gemm<<<gemmGridC, gemmBlk, 0, stream>>>(X1, AGG, Ws2, Wn2, b2,
                                               nullptr, nullptr, nullptr, nullptr, nullptr,
                                               OUT, N, C, 0);

  log_softmax_k<<<(int)((wavesR * 32 + 255) / 256), 256, 0, stream>>>(OUT, N, C);
}